// OrthogonalFusion_76879914598836
// MI455X (gfx1250) — compile-verified
//
#include <hip/hip_runtime.h>

typedef float v2f __attribute__((ext_vector_type(2)));
typedef float v8f __attribute__((ext_vector_type(8)));
typedef unsigned int v4u __attribute__((ext_vector_type(4)));
typedef int v4i __attribute__((ext_vector_type(4)));
typedef int v8i __attribute__((ext_vector_type(8)));

#define C_DIM   1024
#define HW_DIM  1024
#define TILE_HW 32
#define NTHREADS 128

// Dynamic LDS layout (floats):
//   ldsA  [C_DIM][TILE_HW]  : 32768  (128 KB)  staged local tile
//   ldsG  [C_DIM]           : 1024   (4 KB)    g vector for this batch
//   sCoef [TILE_HW]         : 32               coef accumulators
//   sRed  [NTHREADS]        : 128              gsq reduction
//   sGsq  [1]
#define SMEM_FLOATS (C_DIM*TILE_HW + C_DIM + TILE_HW + NTHREADS + 4)

// Probe the builtins ONLY in the device pass; the host pass doesn't see
// AMDGCN builtins and must get plain-C++ placeholder code instead.
#if defined(__HIP_DEVICE_COMPILE__)
#if !__has_builtin(__builtin_amdgcn_tensor_load_to_lds)
#error "DEVICE: tensor_load_to_lds builtin not available"
#endif
#if !__has_builtin(__builtin_amdgcn_s_wait_tensorcnt)
#error "DEVICE: s_wait_tensorcnt builtin not available"
#endif
#if !__has_builtin(__builtin_amdgcn_wmma_f32_16x16x4_f32)
#error "DEVICE: wmma_f32_16x16x4_f32 builtin not available"
#endif
#endif

__global__ void __launch_bounds__(NTHREADS)
OrthogonalFusion_kernel(const float* __restrict__ lf,
                        const float* __restrict__ gf,
                        float* __restrict__ out)
{
    extern __shared__ float smem[];
    float* ldsA  = smem;
    float* ldsG  = smem + C_DIM * TILE_HW;
    float* sCoef = ldsG + C_DIM;
    float* sRed  = sCoef + TILE_HW;
    float* sGsq  = sRed + NTHREADS;

    const int tid  = threadIdx.x;
    const int lane = tid & 31;
    const int wid  = tid >> 5;
    const int b    = blockIdx.y;
    const int hw0  = blockIdx.x * TILE_HW;

    const float* gb = gf + (size_t)b * C_DIM;
    const float* lb = lf + (size_t)b * C_DIM * HW_DIM + hw0;

    // ---- TDM: async-stage local tile [C_DIM x TILE_HW] into LDS ----
#if defined(__HIP_DEVICE_COMPILE__)
    if (wid == 0) {
        unsigned long long ga = (unsigned long long)(uintptr_t)lb;
        unsigned int ldsOff   = (unsigned int)(uintptr_t)ldsA; // low 32 bits = LDS byte offset

        v4u g0;
        g0[0] = 1u;                                            // count=1, user descriptor
        g0[1] = ldsOff;                                        // lds_addr
        g0[2] = (unsigned int)(ga & 0xFFFFFFFFull);            // global_addr[31:0]
        g0[3] = (unsigned int)((ga >> 32) & 0x1FFFFFFull)      // global_addr[56:32]
              | (2u << 30);                                    // type=2 ("image")

        v8i g1;
        g1[0] = (int)(2u << 16);                               // data_size=4B
        g1[1] = (int)((unsigned)HW_DIM << 16);                 // tensor_dim0[15:0]
        g1[2] = (int)((unsigned)C_DIM << 16);                  // dim0[31:16]=0 | tensor_dim1[15:0]
        g1[3] = (int)((unsigned)TILE_HW << 16);                // dim1[31:16]=0 | tile_dim0=32
        g1[4] = C_DIM;                                         // tile_dim1=1024 | tile_dim2=0
        g1[5] = HW_DIM;                                        // tensor_dim0_stride[31:0]
        g1[6] = 0;                                             // stride0 hi | dim1_stride lo
        g1[7] = 0;

        v4i gz = {0, 0, 0, 0};
#if __clang_major__ >= 23
        v8i gz8 = {0, 0, 0, 0, 0, 0, 0, 0};
        __builtin_amdgcn_tensor_load_to_lds(g0, g1, gz, gz, gz8, 0);
#else
        __builtin_amdgcn_tensor_load_to_lds(g0, g1, gz, gz, 0);
#endif
    }
#else
    // host-pass placeholder (never executed on device)
    for (int i = tid; i < (C_DIM * TILE_HW) / 4; i += NTHREADS) {
        int c  = i >> 3;
        int h4 = (i & 7) * 4;
        *(float4*)&ldsA[c * TILE_HW + h4] =
            *(const float4*)&lb[(size_t)c * HW_DIM + h4];
    }
#endif

    // ---- overlap with DMA: stage g into LDS + per-thread gsq partial ----
    float gs = 0.f;
    for (int i = tid; i < C_DIM; i += NTHREADS) {
        float v = gb[i];
        ldsG[i] = v;
        gs += v * v;
    }
    sRed[tid] = gs;
    if (tid < TILE_HW) sCoef[tid] = 0.f;

#if defined(__HIP_DEVICE_COMPILE__)
    if (wid == 0) {
        __builtin_amdgcn_s_wait_tensorcnt(0);
    }
#endif
    __syncthreads();

    if (tid == 0) {
        float s = 0.f;
        for (int i = 0; i < NTHREADS; ++i) s += sRed[i];
        sGsq[0] = s;
    }

    // ---- coef reduction: wave w handles hw-half (w&1), C-half (w>>1) ----
    const int hwb  = (wid & 1) * 16;
    const int c0   = (wid >> 1) * (C_DIM / 2);
    const int col  = hwb + (lane & 15);
    const int rsel = (lane < 16) ? 0 : 2;

#if defined(__HIP_DEVICE_COMPILE__)
    // Two independent accumulator chains so back-to-back WMMAs aren't
    // serialized on the same D->C dependency.
    v8f acc0 = {0.f, 0.f, 0.f, 0.f, 0.f, 0.f, 0.f, 0.f};
    v8f acc1 = {0.f, 0.f, 0.f, 0.f, 0.f, 0.f, 0.f, 0.f};
    for (int c = c0; c < c0 + C_DIM / 2; c += 8) {
        v2f a0, b0, a1, b1;
        // A (16x4 f32): all 16 rows identical = g chunk; lanes0-15 K=0,1; lanes16-31 K=2,3
        a0.x = ldsG[c + rsel];
        a0.y = ldsG[c + rsel + 1];
        // B (4x16 f32): row K, column = lane&15
        b0.x = ldsA[(c + rsel) * TILE_HW + col];
        b0.y = ldsA[(c + rsel + 1) * TILE_HW + col];
        a1.x = ldsG[c + 4 + rsel];
        a1.y = ldsG[c + 4 + rsel + 1];
        b1.x = ldsA[(c + 4 + rsel) * TILE_HW + col];
        b1.y = ldsA[(c + 4 + rsel + 1) * TILE_HW + col];
        acc0 = __builtin_amdgcn_wmma_f32_16x16x4_f32(
                   false, a0, false, b0, (short)0, acc0, false, false);
        acc1 = __builtin_amdgcn_wmma_f32_16x16x4_f32(
                   false, a1, false, b1, (short)0, acc1, false, false);
    }
    float partial = acc0[0] + acc1[0]; // D[m][col] identical over m -> coef partial
#else
    // host-pass placeholder
    float partial = 0.f;
    if (lane < 16)
        for (int c = c0; c < c0 + C_DIM / 2; ++c)
            partial += ldsG[c] * ldsA[c * TILE_HW + col];
#endif

    if (lane < 16) atomicAdd(&sCoef[col], partial);
    __syncthreads();

    // ---- write phase: g broadcast + orthogonal component, float4 stores ----
    const float invg = 1.0f / sGsq[0];
    float* outB = out + (size_t)b * (2 * C_DIM) * HW_DIM + hw0;
    for (int it = 0; it < C_DIM / 16; ++it) {
        int c  = it * 16 + (tid >> 3);
        int h4 = (tid & 7) * 4;
        float4 lv = *(const float4*)&ldsA[c * TILE_HW + h4];
        float4 cf = *(const float4*)&sCoef[h4];
        float  gc = ldsG[c];
        float  s  = gc * invg;
        float4 ov;
        ov.x = lv.x - s * cf.x;
        ov.y = lv.y - s * cf.y;
        ov.z = lv.z - s * cf.z;
        ov.w = lv.w - s * cf.w;
        float4 gm = make_float4(gc, gc, gc, gc);
        *(float4*)&outB[(size_t)c * HW_DIM + h4]           = gm;
        *(float4*)&outB[(size_t)(C_DIM + c) * HW_DIM + h4] = ov;
    }
}

extern "C" void kernel_launch(void* const* d_in, const int* in_sizes, int n_in,
                              void* d_out, int out_size, void* d_ws, size_t ws_size,
                              hipStream_t stream) {
    const float* lf = (const float*)d_in[0];
    const float* gf = (const float*)d_in[1];
    float* out = (float*)d_out;
    const int B = in_sizes[1] / C_DIM;                 // 64
    dim3 grid(HW_DIM / TILE_HW, B);                    // (32, 64)
    dim3 block(NTHREADS);
    size_t shmem = (size_t)SMEM_FLOATS * sizeof(float); // ~133 KB
    (void)hipFuncSetAttribute((const void*)OrthogonalFusion_kernel,
                              hipFuncAttributeMaxDynamicSharedMemorySize,
                              (int)shmem);
    OrthogonalFusion_kernel<<<grid, block, shmem, stream>>>(lf, gf, out);
}